// SolarModel_15539191677322
// MI455X (gfx1250) — compile-verified
//
#include <hip/hip_runtime.h>

typedef __attribute__((ext_vector_type(2))) float v2f;
typedef __attribute__((ext_vector_type(8))) float v8f;

#define GCONST 0.000295912208286f

__device__ __forceinline__ float lane_bcast(float v, int srcLane) {
  return __int_as_float(__builtin_amdgcn_readlane(__float_as_int(v), srcLane));
}

__device__ __forceinline__ v8f wmma_k4(v2f a, v2f b, v8f c) {
  return __builtin_amdgcn_wmma_f32_16x16x4_f32(false, a, false, b, (short)0, c,
                                               false, false);
}

// p[5] at runtime index i (0..4); 0 otherwise (sun at origin / padding rows)
__device__ __forceinline__ float sel5(const float p[5], int i) {
  float r = 0.0f;
  r = (i == 0) ? p[0] : r;
  r = (i == 1) ? p[1] : r;
  r = (i == 2) ? p[2] : r;
  r = (i == 3) ? p[3] : r;
  r = (i == 4) ? p[4] : r;
  return r;
}

// acc(5x3) via D = Aoff(16x6, diag=0) x B(6x4) using two chained
// V_WMMA_F32_16X16X4_F32 ops. B columns 0..2 = positions, column 3 = ones,
// so D[i][3] = rowsum_i and acc_i = D[i][0..2] - rowsum_i * pos_i, which is
// exactly sum_j w_ij (pos_j - pos_i), the reference force law (sun at origin).
//
// Column->K mapping (balances work across wave halves; 3 rsq chains/lane):
//   WMMA#1: K=0..3 <-> bodies 0..3 (low half K=0,1; high half K=2,3)
//   WMMA#2: K'=0 <-> body4 (low half), K'=2 <-> body5=sun (high half),
//           K'=1,3 unused (zero).
__device__ __forceinline__ void accel(const float px[5], const float py[5],
                                      const float pz[5], const float Gm[6],
                                      int lane, float ax[5], float ay[5],
                                      float az[5]) {
  const int mrow = lane & 15;  // A-matrix row owned by this lane
  const bool hi = lane >= 16;  // high half supplies K=2,3 (and K'=2,3)
  const int ncol = lane & 15;  // B/D column owned by this lane

  const float pmx = sel5(px, mrow);
  const float pmy = sel5(py, mrow);
  const float pmz = sel5(pz, mrow);

  // Per-lane slot -> target body: {hi?2:0, hi?3:1, hi?5(sun):4}
  float qx[3], qy[3], qz[3], Gms[3];
  qx[0] = hi ? px[2] : px[0]; qy[0] = hi ? py[2] : py[0]; qz[0] = hi ? pz[2] : pz[0];
  qx[1] = hi ? px[3] : px[1]; qy[1] = hi ? py[3] : py[1]; qz[1] = hi ? pz[3] : pz[1];
  qx[2] = hi ? 0.0f  : px[4]; qy[2] = hi ? 0.0f  : py[4]; qz[2] = hi ? 0.0f  : pz[4];
  Gms[0] = hi ? Gm[2] : Gm[0];
  Gms[1] = hi ? Gm[3] : Gm[1];
  Gms[2] = hi ? Gm[5] : Gm[4];

  float wv[3];
#pragma unroll
  for (int s = 0; s < 3; ++s) {
    const float dx = pmx - qx[s];
    const float dy = pmy - qy[s];
    const float dz = pmz - qz[s];
    const float r2 = dx * dx + dy * dy + dz * dz;
    const float inv = __builtin_amdgcn_rsqf(r2);  // v_rsq_f32
    wv[s] = Gms[s] * (inv * inv) * inv;           // G*m_j / r^3
  }

  // Zero masks (all loop-invariant lane predicates -> SGPR masks):
  const bool rowValid = (mrow < 5);              // drop sun row + pad rows
  const int kb0 = hi ? 2 : 0;                    // slot0/slot1 bodies
  const int kb2 = hi ? 5 : 4;                    // slot2 body
  const bool z0 = !rowValid || (kb0 == mrow);    // diagonal -> 0
  const bool z1 = !rowValid || (kb0 + 1 == mrow);
  const bool z2 = !rowValid || (kb2 == mrow);

  v2f afrag0, afrag1, bfrag0, bfrag1;
  afrag0.x = z0 ? 0.0f : wv[0];
  afrag0.y = z1 ? 0.0f : wv[1];
  afrag1.x = z2 ? 0.0f : wv[2];
  afrag1.y = 0.0f;  // unused columns K'=1 / K'=3

  // B entry for slot s at column ncol: positions for n<3, ones column at n==3
  const bool n0 = (ncol == 0), n1 = (ncol == 1), n2 = (ncol == 2), n3 = (ncol == 3);
  auto Bent = [&](int s) -> float {
    return n0 ? qx[s] : n1 ? qy[s] : n2 ? qz[s] : n3 ? 1.0f : 0.0f;
  };
  bfrag0.x = Bent(0);
  bfrag0.y = Bent(1);
  bfrag1.x = Bent(2);
  bfrag1.y = 0.0f;

  v8f c = {0.0f, 0.0f, 0.0f, 0.0f, 0.0f, 0.0f, 0.0f, 0.0f};
  v8f d = wmma_k4(afrag0, bfrag0, c);
  d = wmma_k4(afrag1, bfrag1, d);

  // D layout: body i -> VGPR i, column n -> lane n. Re-replicate via readlane.
#pragma unroll
  for (int i = 0; i < 5; ++i) {
    const float rs = lane_bcast(d[i], 3);  // rowsum_i
    ax[i] = fmaf(-rs, px[i], lane_bcast(d[i], 0));
    ay[i] = fmaf(-rs, py[i], lane_bcast(d[i], 1));
    az[i] = fmaf(-rs, pz[i], lane_bcast(d[i], 2));
  }
}

__global__ void __launch_bounds__(32)
solar_rk4_kernel(const float* __restrict__ param, const int* __restrict__ pTinit,
                 const int* __restrict__ pTfin, const int* __restrict__ pDt,
                 float* __restrict__ out, int outW) {
  const int lane = threadIdx.x;
  const int dti = *pDt;
  const int N = (*pTfin) / dti;
  const int Ninit = (*pTinit) / dti;
  const float dt = (float)dti;
  const float h2 = 0.5f * dt;
  const float h6 = dt / 6.0f;

  float Gm[6];
#pragma unroll
  for (int j = 0; j < 6; ++j) Gm[j] = GCONST * fabsf(param[j]);  // sqrt(m^2+1e-64)==|m| in f32

  float px[5], py[5], pz[5], vx[5], vy[5], vz[5];
#pragma unroll
  for (int i = 0; i < 5; ++i) {
    px[i] = param[6 + 3 * i + 0];
    py[i] = param[6 + 3 * i + 1];
    pz[i] = param[6 + 3 * i + 2];
    vx[i] = param[21 + 3 * i + 0];
    vy[i] = param[21 + 3 * i + 1];
    vz[i] = param[21 + 3 * i + 2];
  }

  // Output mapping (loop-invariant): lane r -> planet r/3, component r%3
  const int prow = lane / 3;
  const int pcomp = lane - 3 * prow;
  const bool pc0 = (pcomp == 0), pc1 = (pcomp == 1);
  const bool storeLane = (lane < 15);
  float* const outp = out + lane * outW;

  float ax[5], ay[5], az[5];
  float spx[5], spy[5], spz[5], svx[5], svy[5], svz[5];
  float qx[5], qy[5], qz[5];  // staged positions
  float wx[5], wy[5], wz[5];  // k2 / k4 velocity parts
  float ux[5], uy[5], uz[5];  // k3 velocity part

  for (int t = 1; t <= N; ++t) {
    // k1
    accel(px, py, pz, Gm, lane, ax, ay, az);
#pragma unroll
    for (int i = 0; i < 5; ++i) {
      spx[i] = vx[i]; spy[i] = vy[i]; spz[i] = vz[i];
      svx[i] = ax[i]; svy[i] = ay[i]; svz[i] = az[i];
      qx[i] = fmaf(h2, vx[i], px[i]);
      qy[i] = fmaf(h2, vy[i], py[i]);
      qz[i] = fmaf(h2, vz[i], pz[i]);
      wx[i] = fmaf(h2, ax[i], vx[i]);
      wy[i] = fmaf(h2, ay[i], vy[i]);
      wz[i] = fmaf(h2, az[i], vz[i]);
    }
    // k2
    accel(qx, qy, qz, Gm, lane, ax, ay, az);
#pragma unroll
    for (int i = 0; i < 5; ++i) {
      spx[i] = fmaf(2.0f, wx[i], spx[i]); spy[i] = fmaf(2.0f, wy[i], spy[i]); spz[i] = fmaf(2.0f, wz[i], spz[i]);
      svx[i] = fmaf(2.0f, ax[i], svx[i]); svy[i] = fmaf(2.0f, ay[i], svy[i]); svz[i] = fmaf(2.0f, az[i], svz[i]);
      qx[i] = fmaf(h2, wx[i], px[i]);
      qy[i] = fmaf(h2, wy[i], py[i]);
      qz[i] = fmaf(h2, wz[i], pz[i]);
      ux[i] = fmaf(h2, ax[i], vx[i]);
      uy[i] = fmaf(h2, ay[i], vy[i]);
      uz[i] = fmaf(h2, az[i], vz[i]);
    }
    // k3
    accel(qx, qy, qz, Gm, lane, ax, ay, az);
#pragma unroll
    for (int i = 0; i < 5; ++i) {
      spx[i] = fmaf(2.0f, ux[i], spx[i]); spy[i] = fmaf(2.0f, uy[i], spy[i]); spz[i] = fmaf(2.0f, uz[i], spz[i]);
      svx[i] = fmaf(2.0f, ax[i], svx[i]); svy[i] = fmaf(2.0f, ay[i], svy[i]); svz[i] = fmaf(2.0f, az[i], svz[i]);
      qx[i] = fmaf(dt, ux[i], px[i]);
      qy[i] = fmaf(dt, uy[i], py[i]);
      qz[i] = fmaf(dt, uz[i], pz[i]);
      wx[i] = fmaf(dt, ax[i], vx[i]);
      wy[i] = fmaf(dt, ay[i], vy[i]);
      wz[i] = fmaf(dt, az[i], vz[i]);
    }
    // k4
    accel(qx, qy, qz, Gm, lane, ax, ay, az);
#pragma unroll
    for (int i = 0; i < 5; ++i) {
      spx[i] += wx[i]; spy[i] += wy[i]; spz[i] += wz[i];
      svx[i] += ax[i]; svy[i] += ay[i]; svz[i] += az[i];
      px[i] = fmaf(h6, spx[i], px[i]); py[i] = fmaf(h6, spy[i], py[i]); pz[i] = fmaf(h6, spz[i], pz[i]);
      vx[i] = fmaf(h6, svx[i], vx[i]); vy[i] = fmaf(h6, svy[i], vy[i]); vz[i] = fmaf(h6, svz[i], vz[i]);
    }

    // Branchless value select (all lanes), minimal divergence on the store only
    const float vpx = sel5(px, prow);
    const float vpy = sel5(py, prow);
    const float vpz = sel5(pz, prow);
    const float val = pc0 ? vpx : (pc1 ? vpy : vpz);
    if (storeLane & (t >= Ninit)) {
      __builtin_nontemporal_store(val, outp + (t - Ninit));
    }
  }
}

extern "C" void kernel_launch(void* const* d_in, const int* in_sizes, int n_in,
                              void* d_out, int out_size, void* d_ws, size_t ws_size,
                              hipStream_t stream) {
  const float* param = (const float*)d_in[0];
  const int* tinit = (const int*)d_in[1];
  const int* tfin = (const int*)d_in[2];
  const int* dt = (const int*)d_in[3];
  const int outW = out_size / 15;  // = Tfin/dt - Tinit/dt + 1
  solar_rk4_kernel<<<1, 32, 0, stream>>>(param, tinit, tfin, dt, (float*)d_out,
                                         outW);
}